// mamba2_time_series_89532888252834
// MI455X (gfx1250) — compile-verified
//
#include <hip/hip_runtime.h>
#include <math.h>

// ---------------- model constants (match reference) ----------------
#define BSZ   2
#define SL    1024
#define DM    768
#define DI    1536
#define NH    24
#define HD    64
#define DS    128
#define DC    4
#define CD    (DI + 2*DS)            // 1792
#define NL    24
#define DPROJ (2*DI + 2*DS + NH)     // 3352
#define QC    64                     // chunk length
#define NC    (SL/QC)                // 16 chunks
#define FEPS  1e-5f

typedef __attribute__((ext_vector_type(2))) float v2f;
typedef __attribute__((ext_vector_type(4))) float v4f;
typedef __attribute__((ext_vector_type(8))) float v8f;
typedef __attribute__((ext_vector_type(4))) int   v4i;

#define AS1 __attribute__((address_space(1)))
#define AS3 __attribute__((address_space(3)))

#if defined(__has_builtin)
#  if __has_builtin(__builtin_amdgcn_global_load_async_to_lds_b128)
#    define HAVE_ASYNC_BUILTIN 1
#  endif
#  if __has_builtin(__builtin_amdgcn_s_wait_asynccnt)
#    define HAVE_WAIT_BUILTIN 1
#  endif
#endif

// 16-byte async copy global -> LDS (ASYNCcnt-tracked, no VGPR round trip)
__device__ __forceinline__ void async_copy16(float* lds, const float* g)
{
#ifdef HAVE_ASYNC_BUILTIN
    __builtin_amdgcn_global_load_async_to_lds_b128(
        (AS1 v4i*)(g), (AS3 v4i*)(lds), 0, 0);
#else
    unsigned loff = (unsigned)(unsigned long long)(AS3 void*)lds;
    asm volatile("global_load_async_to_lds_b128 %0, %1, off"
                 :: "v"(loff), "v"((unsigned long long)g) : "memory");
#endif
}
__device__ __forceinline__ void wait_asynccnt_le3()
{
#ifdef HAVE_WAIT_BUILTIN
    __builtin_amdgcn_s_wait_asynccnt(3);
#else
    asm volatile("s_wait_asynccnt 0x3" ::: "memory");
#endif
}
__device__ __forceinline__ void wait_asynccnt_le0()
{
#ifdef HAVE_WAIT_BUILTIN
    __builtin_amdgcn_s_wait_asynccnt(0);
#else
    asm volatile("s_wait_asynccnt 0x0" ::: "memory");
#endif
}

// =====================================================================
// Generic batched WMMA fp32 GEMM:  C[M,N] = A[M,K] * B[N,K]^T  (+C if accum)
// A row-major (lda), B row-major (ldb), C row-major (ldc).
// Block tile 64x32, 8 wave32 waves, v_wmma_f32_16x16x4_f32, double-buffered
// LDS staged via async global->LDS b128 copies.
// Requires M % 64 == 0, K % 32 == 0; rows 16B-aligned (lda/ldb % 4 == 0).
// N may be ragged (clamped async loads + zero-fix, guarded stores).
// =====================================================================
__global__ __launch_bounds__(256) void gemm_wmma_f32(
    const float* __restrict__ A, const float* __restrict__ Bm, float* __restrict__ C,
    int M, int N, int K, int lda, int ldb, int ldc,
    long long sA, long long sB, long long sC, int accum)
{
    A  += (long long)blockIdx.z * sA;
    Bm += (long long)blockIdx.z * sB;
    C  += (long long)blockIdx.z * sC;

    // row stride 36 floats: 144 B (16B aligned for b128) and conflict-free
    // (36*lane mod 64 distinct over 16 lanes)
    __shared__ alignas(16) float As[2][64 * 36];
    __shared__ alignas(16) float Bs[2][32 * 36];

    const int tid  = threadIdx.x;
    const int lane = tid & 31;
    const int w    = tid >> 5;
    const int wm   = w & 3;        // 4 wave-tiles along M
    const int wn   = w >> 2;       // 2 wave-tiles along N
    const int half = lane >> 4;    // 0: K pair {0,1}, 1: K pair {2,3}
    const int l16  = lane & 15;
    const int bm   = blockIdx.y;
    const int bn   = blockIdx.x;

    // staging coords: thread -> one b128 of B, two b128 of A per K-tile
    const int r0  = tid >> 3;            // 0..31
    const int c4  = (tid & 7) * 4;       // 0,4,...,28
    const int gnB = bn * 32 + r0;
    const long long rowA0 = (long long)(bm * 64 + r0) * lda;
    const long long rowA1 = (long long)(bm * 64 + r0 + 32) * lda;
    const long long rowB  = (long long)((gnB < N) ? gnB : (N - 1)) * ldb;

    const int T = K >> 5;
    v8f acc = {};

    // prologue: issue K-tile 0 into buffer 0
    async_copy16(&As[0][r0 * 36 + c4],        A + rowA0 + c4);
    async_copy16(&As[0][(r0 + 32) * 36 + c4], A + rowA1 + c4);
    async_copy16(&Bs[0][r0 * 36 + c4],        Bm + rowB + c4);

    for (int t = 0; t < T; ++t) {
        const int buf = t & 1;
        if (t + 1 < T) {
            const int k1 = (t + 1) << 5;
            const int nb = buf ^ 1;
            async_copy16(&As[nb][r0 * 36 + c4],        A + rowA0 + k1 + c4);
            async_copy16(&As[nb][(r0 + 32) * 36 + c4], A + rowA1 + k1 + c4);
            async_copy16(&Bs[nb][r0 * 36 + c4],        Bm + rowB + k1 + c4);
            wait_asynccnt_le3();   // tile t landed; tile t+1 (3 ops) in flight
        } else {
            wait_asynccnt_le0();
        }
        if (gnB >= N) *(v4f*)&Bs[buf][r0 * 36 + c4] = (v4f){0.f, 0.f, 0.f, 0.f};
        __syncthreads();

        const float* ar = &As[buf][(wm * 16 + l16) * 36];
        const float* br = &Bs[buf][(wn * 16 + l16) * 36];
        #pragma unroll
        for (int kk = 0; kk < 32; kk += 4) {
            v2f a, b;
            a.x = ar[kk + 2 * half];
            a.y = ar[kk + 2 * half + 1];
            b.x = br[kk + 2 * half];
            b.y = br[kk + 2 * half + 1];
            acc = __builtin_amdgcn_wmma_f32_16x16x4_f32(
                      false, a, false, b, (short)0, acc, false, false);
        }
        __syncthreads();
    }

    // C/D layout: vgpr j, lanes 0-15 -> M=j, N=l16 ; lanes 16-31 -> M=j+8
    const int gn = bn * 32 + wn * 16 + l16;
    if (gn < N) {
        const int gm0 = bm * 64 + wm * 16 + 8 * half;
        #pragma unroll
        for (int j = 0; j < 8; ++j) {
            long long idx = (long long)(gm0 + j) * ldc + gn;
            float v = acc[j];
            C[idx] = accum ? (C[idx] + v) : v;
        }
    }
}

// =====================================================================
// helpers
// =====================================================================
__device__ __forceinline__ float gelu_exact(float v) {
    return 0.5f * v * (1.f + erff(v * 0.70710678118654752f));
}
__device__ __forceinline__ float silu(float v) {
    return v / (1.f + expf(-v));
}

// rmsnorm over last dim D, per-row block
__global__ __launch_bounds__(256) void rmsnorm_kernel(
    const float* __restrict__ x, const float* __restrict__ w,
    float* __restrict__ out, int D)
{
    const int row = blockIdx.x;
    const float* xr = x + (long long)row * D;
    float* orow = out + (long long)row * D;
    __shared__ float red[256];
    float s = 0.f;
    for (int d = threadIdx.x; d < D; d += 256) { float v = xr[d]; s += v * v; }
    red[threadIdx.x] = s; __syncthreads();
    for (int off = 128; off > 0; off >>= 1) {
        if (threadIdx.x < off) red[threadIdx.x] += red[threadIdx.x + off];
        __syncthreads();
    }
    const float inv = rsqrtf(red[0] / (float)D + FEPS);
    for (int d = threadIdx.x; d < D; d += 256) orow[d] = xr[d] * inv * w[d];
}

// dt = softplus(zx[..., DI+CD+h] + dtb[h])
__global__ void dt_kernel(const float* __restrict__ zx,
                          const float* __restrict__ dtb,
                          float* __restrict__ dt)
{
    int i = blockIdx.x * 256 + threadIdx.x;         // B*L*NH
    if (i >= BSZ * SL * NH) return;
    int h = i % NH; int bl = i / NH;
    float v = zx[(long long)bl * DPROJ + DI + CD + h] + dtb[h];
    dt[i] = (v > 20.f) ? v : log1pf(expf(v));
}

// causal depthwise conv (k=4) + bias + silu over xBC slice of zx
__global__ void convx_kernel(const float* __restrict__ zx,
                             const float* __restrict__ w,
                             const float* __restrict__ bias,
                             float* __restrict__ out)
{
    long long i = (long long)blockIdx.x * 256 + threadIdx.x;   // B*L*CD
    if (i >= (long long)BSZ * SL * CD) return;
    int ch = (int)(i % CD); long long bl = i / CD;
    int l = (int)(bl % SL); long long b = bl / SL;
    float acc = bias[ch];
    #pragma unroll
    for (int j = 0; j < DC; ++j) {
        int ll = l - (DC - 1) + j;
        if (ll >= 0)
            acc += w[ch * DC + j] * zx[((b * SL + ll) * DPROJ) + DI + ch];
    }
    out[i] = silu(acc);
}

// per-chunk inclusive cumsum of a = dt * (-exp(alog))
__global__ void cumsum_kernel(const float* __restrict__ dt,
                              const float* __restrict__ alog,
                              float* __restrict__ cs)
{
    int i = blockIdx.x * 256 + threadIdx.x;        // B*NH*NC
    if (i >= BSZ * NH * NC) return;
    int c = i % NC; int bh = i / NC;
    int h = bh % NH; int b = bh / NH;
    const float A = -expf(alog[h]);
    float run = 0.f;
    for (int q = 0; q < QC; ++q) {
        int l = c * QC + q;
        run += dt[((long long)(b * SL + l)) * NH + h] * A;
        cs[((long long)(b * NH + h)) * SL + l] = run;
    }
}

// per-chunk states[b,c,h,p,n] = sum_l B[l,n] * exp(csLast - cs[l]) * dt[l] * xh[l,p]
__global__ __launch_bounds__(256) void states_kernel(
    const float* __restrict__ xbc, const float* __restrict__ dt,
    const float* __restrict__ cs, float* __restrict__ st)
{
    const int blk = blockIdx.x;                    // b*NC*NH
    const int h = blk % NH; const int bc = blk / NH;
    const int c = bc % NC;  const int b  = bc / NC;
    __shared__ float xs[QC * HD];
    __shared__ float csL[QC];
    const int tid = threadIdx.x;
    if (tid < QC) csL[tid] = cs[((long long)(b * NH + h)) * SL + c * QC + tid];
    __syncthreads();
    const float csLast = csL[QC - 1];
    for (int i = tid; i < QC * HD; i += 256) {
        int l = i >> 6, p = i & 63;
        float e = expf(csLast - csL[l]) *
                  dt[((long long)(b * SL + c * QC + l)) * NH + h];
        xs[i] = xbc[((long long)(b * SL + c * QC + l)) * CD + h * HD + p] * e;
    }
    __syncthreads();
    const int n  = tid & 127;
    const int pg = tid >> 7;                       // 0..1
    float acc[32];
    #pragma unroll
    for (int j = 0; j < 32; ++j) acc[j] = 0.f;
    const float* Bc = xbc + ((long long)(b * SL + c * QC)) * CD + DI;
    for (int l = 0; l < QC; ++l) {
        float bb = Bc[(long long)l * CD + n];
        #pragma unroll
        for (int j = 0; j < 32; ++j) acc[j] += bb * xs[l * HD + pg * 32 + j];
    }
    float* so = st + ((long long)((b * NC + c) * NH + h)) * HD * DS;
    for (int j = 0; j < 32; ++j) so[(long long)(pg * 32 + j) * DS + n] = acc[j];
}

// sequential inter-chunk recurrence: rewrite st[c] := state entering chunk c
__global__ void state_recur_kernel(const float* __restrict__ cs,
                                   float* __restrict__ st)
{
    long long i = (long long)blockIdx.x * 256 + threadIdx.x;   // B*NH*HD*DS
    if (i >= (long long)BSZ * NH * HD * DS) return;
    int n = (int)(i % DS); long long t = i / DS;
    int p = (int)(t % HD); t /= HD;
    int h = (int)(t % NH); int b = (int)(t / NH);
    float P = 0.f;
    for (int c = 0; c < NC; ++c) {
        long long idx = (((long long)((b * NC + c) * NH + h)) * HD + p) * DS + n;
        float cur = st[idx];
        st[idx] = P;                                   // prev entering chunk c
        float Asum = cs[((long long)(b * NH + h)) * SL + c * QC + QC - 1];
        P = expf(Asum) * P + cur;
    }
}

// y[l,p] = sum_{s<=l} G[l,s]*exp(cs[l]-cs[s])*dt[s]*xh[s,p]
//        + exp(cs[l]) * sum_n C[l,n]*prev[p,n]  +  D*xh[l,p]
__global__ __launch_bounds__(256) void ydiag_kernel(
    const float* __restrict__ xbc, const float* __restrict__ dt,
    const float* __restrict__ cs, const float* __restrict__ G,
    const float* __restrict__ st, const float* __restrict__ dD,
    float* __restrict__ y)
{
    const int blk = blockIdx.x;                    // b*NC*NH
    const int h = blk % NH; const int bc = blk / NH;
    const int c = bc % NC;  const int b  = bc / NC;
    __shared__ float csL[QC];
    __shared__ float dtL[QC];
    const int tid = threadIdx.x;
    if (tid < QC) {
        csL[tid] = cs[((long long)(b * NH + h)) * SL + c * QC + tid];
        dtL[tid] = dt[((long long)(b * SL + c * QC + tid)) * NH + h];
    }
    __syncthreads();
    const int l  = tid >> 2;                       // 0..63
    const int pg = tid & 3;                        // p = pg*16 + j
    const float* xh = xbc + ((long long)(b * SL + c * QC)) * CD + h * HD;
    const float* Cc = xbc + ((long long)(b * SL + c * QC)) * CD + DI + DS;
    const float* Gc = G + ((long long)(b * NC + c)) * QC * QC;
    const float* pv = st + ((long long)((b * NC + c) * NH + h)) * HD * DS;
    const float D = dD[h];
    float acc[16];
    #pragma unroll
    for (int j = 0; j < 16; ++j)
        acc[j] = D * xh[(long long)l * CD + pg * 16 + j];
    const float cl = csL[l];
    for (int s = 0; s <= l; ++s) {
        float wgt = Gc[l * QC + s] * expf(cl - csL[s]) * dtL[s];
        const float* xr = xh + (long long)s * CD + pg * 16;
        #pragma unroll
        for (int j = 0; j < 16; ++j) acc[j] += wgt * xr[j];
    }
    const float ecl = expf(cl);
    for (int n = 0; n < DS; ++n) {
        float cv = Cc[(long long)l * CD + n] * ecl;
        #pragma unroll
        for (int j = 0; j < 16; ++j)
            acc[j] += cv * pv[(long long)(pg * 16 + j) * DS + n];
    }
    float* yo = y + ((long long)(b * SL + c * QC + l)) * DI + h * HD + pg * 16;
    #pragma unroll
    for (int j = 0; j < 16; ++j) yo[j] = acc[j];
}

// out = rmsnorm(y * silu(z), gnw)   (in-place capable: g cached in LDS)
__global__ __launch_bounds__(256) void gatenorm_kernel(
    const float* __restrict__ y, const float* __restrict__ zx,
    const float* __restrict__ gnw, float* __restrict__ out)
{
    const int row = blockIdx.x;                    // B*L
    __shared__ float g[DI];
    __shared__ float red[256];
    const float* yr = y + (long long)row * DI;
    const float* zr = zx + (long long)row * DPROJ; // z = first DI of zxbcdt
    float s = 0.f;
    for (int d = threadIdx.x; d < DI; d += 256) {
        float z = zr[d];
        float v = yr[d] * silu(z);
        g[d] = v; s += v * v;
    }
    red[threadIdx.x] = s; __syncthreads();
    for (int off = 128; off > 0; off >>= 1) {
        if (threadIdx.x < off) red[threadIdx.x] += red[threadIdx.x + off];
        __syncthreads();
    }
    const float inv = rsqrtf(red[0] / (float)DI + FEPS);
    for (int d = threadIdx.x; d < DI; d += 256)
        out[(long long)row * DI + d] = g[d] * inv * gnw[d];
}

// ---------------- frontend ----------------
__global__ void conv1_kernel(const float* __restrict__ x, const float* __restrict__ mask,
                             const float* __restrict__ w, const float* __restrict__ bias,
                             const float* __restrict__ g, const float* __restrict__ bta,
                             const float* __restrict__ m, const float* __restrict__ var,
                             float* __restrict__ out)
{
    int i = blockIdx.x * 256 + threadIdx.x;        // B*384*SL
    if (i >= BSZ * 384 * SL) return;
    int l = i % SL; int t = i / SL;
    int oc = t % 384; int b = t / 384;
    float acc = bias[oc];
    for (int ic = 0; ic < 13; ++ic) {
        const float* src = (ic < 12) ? (x + ((long long)(b * 12 + ic)) * SL)
                                     : (mask + (long long)b * SL);
        #pragma unroll
        for (int j = 0; j < 5; ++j) {
            int ll = l - 2 + j;
            if (ll >= 0 && ll < SL) acc += w[(oc * 13 + ic) * 5 + j] * src[ll];
        }
    }
    acc = gelu_exact(acc);
    out[i] = (acc - m[oc]) * (g[oc] * rsqrtf(var[oc] + FEPS)) + bta[oc];
}

__global__ void conv2_kernel(const float* __restrict__ h1,
                             const float* __restrict__ w, const float* __restrict__ bias,
                             const float* __restrict__ g, const float* __restrict__ bta,
                             const float* __restrict__ m, const float* __restrict__ var,
                             float* __restrict__ hout)
{
    int i = blockIdx.x * 256 + threadIdx.x;        // B*DM*SL, l fastest
    if (i >= BSZ * DM * SL) return;
    int l = i % SL; int t = i / SL;
    int oc = t % DM; int b = t / DM;
    float acc = bias[oc];
    const float* hb = h1 + (long long)b * 384 * SL;
    const float* wr = w + (long long)oc * 384 * 3;
    for (int ic = 0; ic < 384; ++ic) {
        const float* src = hb + (long long)ic * SL;
        #pragma unroll
        for (int j = 0; j < 3; ++j) {
            int ll = l - 1 + j;
            if (ll >= 0 && ll < SL) acc += wr[ic * 3 + j] * src[ll];
        }
    }
    acc = gelu_exact(acc);
    acc = (acc - m[oc]) * (g[oc] * rsqrtf(var[oc] + FEPS)) + bta[oc];
    hout[((long long)(b * SL + l)) * DM + oc] = acc;   // store transposed (B,L,DM)
}

// ---------------- head ----------------
__global__ __launch_bounds__(256) void inv_kernel(const float* __restrict__ h,
                                                  float* __restrict__ invb)
{
    const int row = blockIdx.x;
    __shared__ float red[256];
    const float* xr = h + (long long)row * DM;
    float s = 0.f;
    for (int d = threadIdx.x; d < DM; d += 256) { float v = xr[d]; s += v * v; }
    red[threadIdx.x] = s; __syncthreads();
    for (int off = 128; off > 0; off >>= 1) {
        if (threadIdx.x < off) red[threadIdx.x] += red[threadIdx.x + off];
        __syncthreads();
    }
    if (threadIdx.x == 0) invb[row] = rsqrtf(red[0] / (float)DM + FEPS);
}

__global__ void rep_kernel(const float* __restrict__ h, const float* __restrict__ invb,
                           const float* __restrict__ nfw, float* __restrict__ rep)
{
    int i = blockIdx.x * 256 + threadIdx.x;        // B*DM
    if (i >= BSZ * DM) return;
    int d = i % DM; int b = i / DM;
    float s = 0.f;
    for (int l = 0; l < SL; ++l)
        s += h[((long long)(b * SL + l)) * DM + d] * invb[b * SL + l];
    rep[i] = s * nfw[d] * (1.f / (float)SL);
}

__global__ void head1_kernel(const float* __restrict__ rep, const float* __restrict__ w,
                             const float* __restrict__ bias,
                             const float* __restrict__ g, const float* __restrict__ bta,
                             const float* __restrict__ m, const float* __restrict__ var,
                             float* __restrict__ r)
{
    int i = blockIdx.x * 256 + threadIdx.x;        // B*512
    if (i >= BSZ * 512) return;
    int j = i % 512; int b = i / 512;
    float s = bias[j];
    for (int d = 0; d < DM; ++d) s += rep[b * DM + d] * w[j * DM + d];
    s = (s - m[j]) * (g[j] * rsqrtf(var[j] + FEPS)) + bta[j];
    r[i] = fmaxf(s, 0.f);
}

__global__ void head2_kernel(const float* __restrict__ r, const float* __restrict__ w,
                             const float* __restrict__ bias, float* __restrict__ out)
{
    int i = blockIdx.x * 64 + threadIdx.x;         // B*27
    if (i >= BSZ * 27) return;
    int o = i % 27; int b = i / 27;
    float s = bias[o];
    for (int j = 0; j < 512; ++j) s += r[b * 512 + j] * w[o * 512 + j];
    out[i] = s;
}

// =====================================================================
extern "C" void kernel_launch(void* const* d_in, const int* in_sizes, int n_in,
                              void* d_out, int out_size, void* d_ws, size_t ws_size,
                              hipStream_t stream)
{
    (void)in_sizes; (void)n_in; (void)out_size; (void)ws_size;
    const float* x    = (const float*)d_in[0];
    const float* mask = (const float*)d_in[1];
    const float* c1w  = (const float*)d_in[2];
    const float* c1b  = (const float*)d_in[3];
    const float* bn1g = (const float*)d_in[4];
    const float* bn1b = (const float*)d_in[5];
    const float* bn1m = (const float*)d_in[6];
    const float* bn1v = (const float*)d_in[7];
    const float* c2w  = (const float*)d_in[8];
    const float* c2b  = (const float*)d_in[9];
    const float* bn2g = (const float*)d_in[10];
    const float* bn2b = (const float*)d_in[11];
    const float* bn2m = (const float*)d_in[12];
    const float* bn2v = (const float*)d_in[13];
    const float* ln_w = (const float*)d_in[14];
    const float* in_w = (const float*)d_in[15];
    const float* cw   = (const float*)d_in[16];
    const float* cb   = (const float*)d_in[17];
    const float* dtb  = (const float*)d_in[18];
    const float* alog = (const float*)d_in[19];
    const float* dDp  = (const float*)d_in[20];
    const float* gnw  = (const float*)d_in[21];
    const float* outw = (const float*)d_in[22];
    const float* nfw  = (const float*)d_in[23];
    const float* f1w  = (const float*)d_in[24];
    const float* f1b  = (const float*)d_in[25];
    const float* bn3g = (const float*)d_in[26];
    const float* bn3b = (const float*)d_in[27];
    const float* bn3m = (const float*)d_in[28];
    const float* bn3v = (const float*)d_in[29];
    const float* f2w  = (const float*)d_in[30];
    const float* f2b  = (const float*)d_in[31];

    float* ws = (float*)d_ws;
    float* h    = ws; ws += (size_t)BSZ * SL * DM;
    float* u    = ws; ws += (size_t)BSZ * SL * DM;
    float* zx   = ws; ws += (size_t)BSZ * SL * DPROJ;
    float* xbc  = ws; ws += (size_t)BSZ * SL * CD;
    float* dt   = ws; ws += (size_t)BSZ * SL * NH;
    float* cs   = ws; ws += (size_t)BSZ * NH * SL;
    float* G    = ws; ws += (size_t)BSZ * NC * QC * QC;
    float* st   = ws; ws += (size_t)BSZ * NC * NH * HD * DS;
    float* y    = ws; ws += (size_t)BSZ * SL * DI;
    float* h1   = ws; ws += (size_t)BSZ * 384 * SL;
    float* invb = ws; ws += (size_t)BSZ * SL;
    float* rep  = ws; ws += (size_t)BSZ * DM;
    float* rbuf = ws; ws += (size_t)BSZ * 512;

    // ---- frontend ----
    conv1_kernel<<<(BSZ * 384 * SL + 255) / 256, 256, 0, stream>>>(
        x, mask, c1w, c1b, bn1g, bn1b, bn1m, bn1v, h1);
    conv2_kernel<<<(BSZ * DM * SL + 255) / 256, 256, 0, stream>>>(
        h1, c2w, c2b, bn2g, bn2b, bn2m, bn2v, h);

    // ---- 24 Mamba-2 layers ----
    for (int i = 0; i < NL; ++i) {
        rmsnorm_kernel<<<BSZ * SL, 256, 0, stream>>>(h, ln_w + (size_t)i * DM, u, DM);

        {   // zx = u @ Wi^T   (2048 x 3352 x 768)
            dim3 grid((DPROJ + 31) / 32, (BSZ * SL) / 64, 1);
            gemm_wmma_f32<<<grid, 256, 0, stream>>>(
                u, in_w + (size_t)i * DPROJ * DM, zx,
                BSZ * SL, DPROJ, DM, DM, DM, DPROJ, 0, 0, 0, 0);
        }

        convx_kernel<<<(int)(((size_t)BSZ * SL * CD + 255) / 256), 256, 0, stream>>>(
            zx, cw + (size_t)i * CD * DC, cb + (size_t)i * CD, xbc);
        dt_kernel<<<(BSZ * SL * NH + 255) / 256, 256, 0, stream>>>(
            zx, dtb + (size_t)i * NH, dt);
        cumsum_kernel<<<(BSZ * NH * NC + 255) / 256, 256, 0, stream>>>(
            dt, alog + (size_t)i * NH, cs);

        {   // G[b,c] = Cc @ Bc^T   (64 x 64 x 128 per chunk, batched over b*NC)
            dim3 grid(2, 1, BSZ * NC);
            gemm_wmma_f32<<<grid, 256, 0, stream>>>(
                xbc + DI + DS, xbc + DI, G,
                QC, QC, DS, CD, CD, QC,
                (long long)QC * CD, (long long)QC * CD, (long long)QC * QC, 0);
        }

        states_kernel<<<BSZ * NC * NH, 256, 0, stream>>>(xbc, dt, cs, st);
        state_recur_kernel<<<(int)(((size_t)BSZ * NH * HD * DS + 255) / 256), 256, 0, stream>>>(cs, st);
        ydiag_kernel<<<BSZ * NC * NH, 256, 0, stream>>>(
            xbc, dt, cs, G, st, dDp + (size_t)i * NH, y);
        gatenorm_kernel<<<BSZ * SL, 256, 0, stream>>>(y, zx, gnw + (size_t)i * DI, y);

        {   // h += y @ Wo^T   (2048 x 768 x 1536, residual accumulate)
            dim3 grid((DM + 31) / 32, (BSZ * SL) / 64, 1);
            gemm_wmma_f32<<<grid, 256, 0, stream>>>(
                y, outw + (size_t)i * DM * DI, h,
                BSZ * SL, DM, DI, DI, DI, DM, 0, 0, 0, 1);
        }
    }

    // ---- head ----
    inv_kernel<<<BSZ * SL, 256, 0, stream>>>(h, invb);
    rep_kernel<<<(BSZ * DM + 255) / 256, 256, 0, stream>>>(h, invb, nfw, rep);
    head1_kernel<<<(BSZ * 512 + 255) / 256, 256, 0, stream>>>(
        rep, f1w, f1b, bn3g, bn3b, bn3m, bn3v, rbuf);
    head2_kernel<<<1, 64, 0, stream>>>(rbuf, f2w, f2b, (float*)d_out);
}